// SSD_59914793779795
// MI455X (gfx1250) — compile-verified
//
#include <hip/hip_runtime.h>
#include <hip/hip_bf16.h>
#include <stdint.h>

// ---------------- problem constants ----------------
#define NP       18854      // priors
#define NB       64         // batch
#define NC       21         // classes incl background
#define NFG      20         // foreground classes
#define TOPK     200
#define PADK     208        // 13 * 16 tile padding
#define NTILE    13
#define NMS_T    0.5f
#define SCORE_T  0.01f

typedef float v2f __attribute__((ext_vector_type(2)));
typedef float v8f __attribute__((ext_vector_type(8)));

// ---------- CDNA5 feature guards (compile-safe either way) ----------
#if defined(__AMDGCN__) && __has_builtin(__builtin_amdgcn_wmma_f32_16x16x4_f32)
#define USE_WMMA 1
#else
#define USE_WMMA 0
#endif

#if defined(__AMDGCN__) && __has_builtin(__builtin_amdgcn_ballot_w32)
#define BALLOT32(x) __builtin_amdgcn_ballot_w32(x)
#else
#define BALLOT32(x) ((uint32_t)__ballot(x))
#endif

// Async global->LDS dword gather (CDNA5 GLOBAL_LOAD_ASYNC_TO_LDS_B32, ASYNCcnt).
// Generic pointer low 32 bits == LDS byte offset per CDNA5 aperture rules.
__device__ __forceinline__ void asyncLoadDwordToLds(const float* g, float* l) {
#if defined(__AMDGCN__)
  unsigned ldsOff = (unsigned)(uintptr_t)l;
  unsigned long long ga = (unsigned long long)(uintptr_t)g;
  asm volatile("global_load_async_to_lds_b32 %0, %1, off"
               :: "v"(ldsOff), "v"(ga) : "memory");
#else
  *l = *g;
#endif
}
__device__ __forceinline__ void waitAsyncZero() {
#if defined(__AMDGCN__)
  asm volatile("s_wait_asynccnt 0" ::: "memory");
#endif
}

// ---------------- kernel 0: zero output ----------------
__global__ void zero_out_kernel(float4* __restrict__ out, int n4) {
  int i = blockIdx.x * blockDim.x + threadIdx.x;
  if (i < n4) out[i] = make_float4(0.f, 0.f, 0.f, 0.f);
}

// ---------------- kernel 1: decode boxes + softmax stats ----------------
// dec[b,p] = [x1,y1,x2,y2]; M[b,p] = row max of conf; R[b,p] = 1/sum(exp(c-M))
__global__ void decode_softmax_kernel(const float* __restrict__ loc,
                                      const float* __restrict__ conf,
                                      const float* __restrict__ priors,
                                      float* __restrict__ dec,
                                      float* __restrict__ M,
                                      float* __restrict__ R) {
  int idx = blockIdx.x * 256 + threadIdx.x;
  if (idx >= NB * NP) return;
  int p = idx % NP;
  float4 l4 = ((const float4*)loc)[idx];
  float4 pr = ((const float4*)priors)[p];
  float cx = pr.x + l4.x * 0.1f * pr.z;
  float cy = pr.y + l4.y * 0.1f * pr.w;
  float w  = pr.z * expf(l4.z * 0.2f);
  float h  = pr.w * expf(l4.w * 0.2f);
  float x1 = cx - w * 0.5f, y1 = cy - h * 0.5f;
  ((float4*)dec)[idx] = make_float4(x1, y1, x1 + w, y1 + h);

  const float* c = conf + (size_t)idx * NC;
  float m = c[0];
#pragma unroll
  for (int k = 1; k < NC; ++k) m = fmaxf(m, c[k]);
  float s = 0.f;
#pragma unroll
  for (int k = 0; k < NC; ++k) s += expf(c[k] - m);
  M[idx] = m;
  R[idx] = 1.0f / s;
}

// ---------------- kernel 2: exact top-200 per (batch, fg class) ----------------
// Radix select on float bits (positive floats -> monotone in uint), jax-style
// tie-break (lowest index first), then bitonic sort of selected 200.
__global__ void topk_kernel(const float* __restrict__ conf,
                            const float* __restrict__ dec,
                            const float* __restrict__ M,
                            const float* __restrict__ R,
                            float* __restrict__ topk) {
  const int inst = blockIdx.x;
  const int b    = inst / NFG;
  const int cls  = inst % NFG + 1;
  const int tid  = threadIdx.x;
  const size_t base = (size_t)b * NP;

  __shared__ uint32_t hist[256];
  __shared__ uint32_t selIdx[256];
  __shared__ uint32_t eqIdx[128];
  __shared__ unsigned long long keys[256];
  __shared__ uint32_t shPrefix, shKneed, shSelAll, shNGt, shNEq, shNSel;

  if (tid == 0) { shPrefix = 0; shKneed = TOPK; shSelAll = 0; shNGt = 0; shNEq = 0; shNSel = 0; }
  __syncthreads();

  auto score = [&](int p) -> float {
    float v = conf[(base + (size_t)p) * NC + cls];
    return expf(v - M[base + p]) * R[base + p];
  };

  // 4 passes x 8-bit digits, MSB first
  for (int pass = 0; pass < 4; ++pass) {
    if (shSelAll) break;                 // uniform (set before last barrier)
    int shift = 24 - 8 * pass;
    hist[tid] = 0;
    __syncthreads();
    uint32_t pref = shPrefix;
    for (int p = tid; p < NP; p += 256) {
      float sc = score(p);
      if (sc > SCORE_T) {
        uint32_t bits = __float_as_uint(sc);
        bool cand = (pass == 0) || ((bits >> (shift + 8)) == (pref >> (shift + 8)));
        if (cand) atomicAdd(&hist[(bits >> shift) & 255u], 1u);
      }
    }
    __syncthreads();
    if (tid == 0) {
      uint32_t need = shKneed, cum = 0; int bsel = -1;
      for (int d = 255; d >= 0; --d) {
        uint32_t c2 = cum + hist[d];
        if (c2 >= need) { bsel = d; break; }
        cum = c2;
      }
      if (bsel < 0) shSelAll = 1;                          // fewer than 200 valid
      else { shPrefix = pref | ((uint32_t)bsel << shift); shKneed = need - cum; }
    }
    __syncthreads();
  }

  // compaction
  const uint32_t T = shPrefix, selAll = shSelAll, KeqNeed = shKneed;
  for (int p = tid; p < NP; p += 256) {
    float sc = score(p);
    if (sc > SCORE_T) {
      uint32_t bits = __float_as_uint(sc);
      if (selAll || bits > T) {
        uint32_t pos = atomicAdd(&shNGt, 1u);
        if (pos < 256) selIdx[pos] = (uint32_t)p;
      } else if (bits == T) {
        uint32_t q = atomicAdd(&shNEq, 1u);
        if (q < 128) eqIdx[q] = (uint32_t)p;               // ties beyond 128: ~prob 0
      }
    }
  }
  __syncthreads();
  if (tid == 0) {
    uint32_t n = shNGt;
    if (!selAll) {
      uint32_t m = shNEq; if (m > 128) m = 128;
      uint32_t take = KeqNeed; if (take > m) take = m;
      for (uint32_t r = 0; r < take; ++r) {                // smallest indices first (jax tie rule)
        uint32_t best = 0xFFFFFFFFu, bi = 0;
        for (uint32_t q = 0; q < m; ++q) { uint32_t v = eqIdx[q]; if (v < best) { best = v; bi = q; } }
        eqIdx[bi] = 0xFFFFFFFFu;
        selIdx[n + r] = best;
      }
      n += take;
    }
    if (n > 256) n = 256;
    shNSel = n;
  }
  __syncthreads();

  // 64-bit keys: ascending sort == descending score, ties ascending index
  uint32_t nSel = shNSel;
  unsigned long long key;
  if (tid < (int)nSel) {
    uint32_t p = selIdx[tid];
    uint32_t bits = __float_as_uint(score((int)p));
    key = ((unsigned long long)(0xFFFFFFFFu - bits) << 32) | p;
  } else key = 0xFFFFFFFFFFFFFFFFull;
  keys[tid] = key;
  __syncthreads();
  for (unsigned kk = 2; kk <= 256; kk <<= 1)
    for (unsigned jj = kk >> 1; jj > 0; jj >>= 1) {
      unsigned i = (unsigned)tid, ixj = i ^ jj;
      if (ixj > i) {
        unsigned long long a = keys[i], b2 = keys[ixj];
        bool up = ((i & kk) == 0);
        if ((a > b2) == up) { keys[i] = b2; keys[ixj] = a; }
      }
      __syncthreads();
    }

  // emit SoA record: [score|x1|y1|x2|y2] x 200
  if (tid < TOPK) {
    float* rec = topk + (size_t)inst * (5 * TOPK);
    if (tid < (int)nSel) {
      unsigned long long k2 = keys[tid];
      uint32_t p    = (uint32_t)(k2 & 0xFFFFFFFFu);
      float    sc   = __uint_as_float(0xFFFFFFFFu - (uint32_t)(k2 >> 32));
      float4   bx   = ((const float4*)dec)[base + p];
      rec[0 * TOPK + tid] = sc;
      rec[1 * TOPK + tid] = bx.x; rec[2 * TOPK + tid] = bx.y;
      rec[3 * TOPK + tid] = bx.z; rec[4 * TOPK + tid] = bx.w;
    } else {
      rec[0 * TOPK + tid] = -1.0f;
      rec[1 * TOPK + tid] = 0.f; rec[2 * TOPK + tid] = 0.f;
      rec[3 * TOPK + tid] = 0.f; rec[4 * TOPK + tid] = 0.f;
    }
  }
}

// ---------------- kernel 3: NMS (WMMA pairwise tiles + bitmask suppression) ----------------
__global__ void nms_kernel(const float* __restrict__ topk, float* __restrict__ out) {
  const int inst = blockIdx.x;
  const int b    = inst / NFG;
  const int cls  = inst % NFG + 1;
  const int tid  = threadIdx.x;
  const int lane = tid & 31;
  const int wv   = tid >> 5;

  __shared__ float sS[PADK], sX1[PADK], sY1[PADK], sX2[PADK], sY2[PADK], sA[PADK];
  __shared__ uint32_t smask[PADK * 7];     // per-row 224-bit "suppresses j" mask
  __shared__ uint32_t svalid[7], skeep[7];

  for (int i = tid; i < PADK * 7; i += 256) smask[i] = 0;
  if (tid < 7) svalid[tid] = 0;

  const float* rec = topk + (size_t)inst * (5 * TOPK);
  if (tid < TOPK) {
    // CDNA5 async global->LDS staging (ASYNCcnt), falls back to plain loads off-arch
    asyncLoadDwordToLds(rec + 0 * TOPK + tid, &sS[tid]);
    asyncLoadDwordToLds(rec + 1 * TOPK + tid, &sX1[tid]);
    asyncLoadDwordToLds(rec + 2 * TOPK + tid, &sY1[tid]);
    asyncLoadDwordToLds(rec + 3 * TOPK + tid, &sX2[tid]);
    asyncLoadDwordToLds(rec + 4 * TOPK + tid, &sY2[tid]);
  } else if (tid < PADK) {
    sS[tid] = -1.f; sX1[tid] = 0.f; sY1[tid] = 0.f; sX2[tid] = 0.f; sY2[tid] = 0.f;
  }
  waitAsyncZero();
  __syncthreads();
  if (tid < PADK) sA[tid] = (sX2[tid] - sX1[tid]) * (sY2[tid] - sY1[tid]);
  if (tid < TOPK && sS[tid] > SCORE_T) atomicOr(&svalid[tid >> 5], 1u << (tid & 31));
  __syncthreads();

  // 13x13 tiles of 16x16; one tile per wave per iteration
  for (int tile = wv; tile < NTILE * NTILE; tile += 8) {
    int ti = tile / NTILE, tj = tile % NTILE;
    int I0 = ti * 16, J0 = tj * 16;
    int n = lane & 15, h = lane >> 4;
    int jg = J0 + n;
    float jx1 = sX1[jg], jy1 = sY1[jg], jx2 = sX2[jg], jy2 = sY2[jg], jar = sA[jg];
    float wj = jx2 - jx1, hj = jy2 - jy1;

    float d1[8], d2[8], d3[8], d4[8], d5[8];   // x1i-x1j, x2i-x2j, y1i-y1j, y2i-y2j, ai+aj
#if USE_WMMA
    {
      // A (16x4 f32): row m = [1, x1, x2, area] (Ax) / [1, y1, y2, 0] (Ay)
      // layout: VGPR0 = K0 (lanes 0-15) | K2 (lanes 16-31); VGPR1 = K1 | K3
      int m = n;
      v2f ax, ay;
      ax.x = h ? sX2[I0 + m] : 1.0f;  ax.y = h ? sA[I0 + m]  : sX1[I0 + m];
      ay.x = h ? sY2[I0 + m] : 1.0f;  ay.y = h ? 0.0f        : sY1[I0 + m];
      v8f z = {0.f, 0.f, 0.f, 0.f, 0.f, 0.f, 0.f, 0.f};
      // B (4x16 f32): col j, same K split across lane halves
      v2f b1; b1.x = h ? 0.f : -jx1; b1.y = h ? 0.f : 1.f;   // [-x1j,1,0,0]
      v2f b2; b2.x = h ? 1.f : -jx2; b2.y = 0.f;             // [-x2j,0,1,0]
      v2f b3; b3.x = h ? 0.f : -jy1; b3.y = h ? 0.f : 1.f;   // [-y1j,1,0,0]
      v2f b4; b4.x = h ? 1.f : -jy2; b4.y = 0.f;             // [-y2j,0,1,0]
      v2f b5; b5.x = h ? 0.f :  jar; b5.y = h ? 1.f : 0.f;   // [aj,0,0,1]
      v8f D1 = __builtin_amdgcn_wmma_f32_16x16x4_f32(false, ax, false, b1, (short)0, z, false, false);
      v8f D2 = __builtin_amdgcn_wmma_f32_16x16x4_f32(false, ax, false, b2, (short)0, z, false, false);
      v8f D3 = __builtin_amdgcn_wmma_f32_16x16x4_f32(false, ay, false, b3, (short)0, z, false, false);
      v8f D4 = __builtin_amdgcn_wmma_f32_16x16x4_f32(false, ay, false, b4, (short)0, z, false, false);
      v8f D5 = __builtin_amdgcn_wmma_f32_16x16x4_f32(false, ax, false, b5, (short)0, z, false, false);
#pragma unroll
      for (int r = 0; r < 8; ++r) { d1[r] = D1[r]; d2[r] = D2[r]; d3[r] = D3[r]; d4[r] = D4[r]; d5[r] = D5[r]; }
    }
#else
#pragma unroll
    for (int r = 0; r < 8; ++r) {
      int ig = I0 + r + 8 * h;
      d1[r] = sX1[ig] - jx1; d2[r] = sX2[ig] - jx2;
      d3[r] = sY1[ig] - jy1; d4[r] = sY2[ig] - jy2;
      d5[r] = sA[ig] + jar;
    }
#endif
    int w = tj >> 1, sh16 = (tj & 1) * 16;
#pragma unroll
    for (int r = 0; r < 8; ++r) {
      int ig = I0 + r + 8 * h;                 // C/D layout: lanes 0-15 -> M=r, 16-31 -> M=r+8
      float iw = fmaxf(wj + fminf(0.f, d2[r]) - fmaxf(0.f, d1[r]), 0.f);
      float ih = fmaxf(hj + fminf(0.f, d4[r]) - fmaxf(0.f, d3[r]), 0.f);
      float inter = iw * ih;
      float uni   = d5[r] - inter;
      bool supp = (inter > NMS_T * fmaxf(uni, 1e-12f)) && (jg > ig);
      uint32_t bal = BALLOT32(supp);           // low16 = row I0+r, high16 = row I0+r+8
      if (lane == 0 && bal) {
        uint32_t lo = bal & 0xFFFFu, hi = bal >> 16;
        if (lo) atomicOr(&smask[(I0 + r)     * 7 + w], lo << sh16);
        if (hi) atomicOr(&smask[(I0 + r + 8) * 7 + w], hi << sh16);
      }
    }
  }
  __syncthreads();

  // serial greedy suppression: keep bits striped over lanes 0..6 of wave 0
  uint32_t keep = (lane < 7) ? svalid[lane] : 0u;
  if (wv == 0) {
    for (int i = 0; i < TOPK; ++i) {
      uint32_t kw = (uint32_t)__shfl((int)keep, i >> 5);
      if ((kw >> (i & 31)) & 1u) {
        uint32_t mr = (lane < 7) ? smask[i * 7 + lane] : 0u;
        keep &= ~mr;
      }
    }
    if (lane < 7) skeep[lane] = keep;
  }
  __syncthreads();

  // compact kept rows (original top-k order) into output
  if (tid < TOPK) {
    uint32_t wq = (uint32_t)tid >> 5, bq = (uint32_t)tid & 31;
    if ((skeep[wq] >> bq) & 1u) {
      uint32_t rank = __popc(skeep[wq] & ((1u << bq) - 1u));
      for (uint32_t q = 0; q < wq; ++q) rank += __popc(skeep[q]);
      float* o = out + (((size_t)b * NC + cls) * TOPK + rank) * 5;
      o[0] = sS[tid]; o[1] = sX1[tid]; o[2] = sY1[tid]; o[3] = sX2[tid]; o[4] = sY2[tid];
    }
  }
}

// ---------------- launcher ----------------
extern "C" void kernel_launch(void* const* d_in, const int* in_sizes, int n_in,
                              void* d_out, int out_size, void* d_ws, size_t ws_size,
                              hipStream_t stream) {
  (void)in_sizes; (void)n_in; (void)ws_size;
  const float* loc    = (const float*)d_in[0];
  const float* conf   = (const float*)d_in[1];
  const float* priors = (const float*)d_in[2];
  float* out = (float*)d_out;

  // workspace layout (~34 MB total)
  char* ws = (char*)d_ws;
  size_t off = 0;
  float* dec  = (float*)(ws + off); off += (size_t)NB * NP * 4 * sizeof(float);
  float* Mx   = (float*)(ws + off); off += (size_t)NB * NP * sizeof(float);
  float* Rx   = (float*)(ws + off); off += (size_t)NB * NP * sizeof(float);
  float* topk = (float*)(ws + off); off += (size_t)NB * NFG * 5 * TOPK * sizeof(float);

  int n4 = out_size / 4;   // 1,344,000 floats, divisible by 4
  zero_out_kernel<<<(n4 + 255) / 256, 256, 0, stream>>>((float4*)out, n4);

  int nElem = NB * NP;
  decode_softmax_kernel<<<(nElem + 255) / 256, 256, 0, stream>>>(loc, conf, priors, dec, Mx, Rx);

  topk_kernel<<<NB * NFG, 256, 0, stream>>>(conf, dec, Mx, Rx, topk);

  nms_kernel<<<NB * NFG, 256, 0, stream>>>(topk, out);
}